// SpikingNeuralNet_4655744549230
// MI455X (gfx1250) — compile-verified
//
#include <hip/hip_runtime.h>
#include <hip/hip_bf16.h>

typedef float v2f __attribute__((ext_vector_type(2)));
typedef float v8f __attribute__((ext_vector_type(8)));

#define T_STEPS 1000
#define S_TOT   64
#define NN      128
#define DT_F    0.01f
#define ROWS    16            // samples per workgroup == WMMA M
#define LDS_STRIDE 130        // 128 + 2 pad: breaks m*128 same-bank pattern

// One persistent kernel: state (v,i,s) lives in registers in the WMMA C/D
// fragment layout; w lives in registers as preloaded B fragments; spikes are
// exchanged through double-buffered LDS once per step.
__global__ __launch_bounds__(256)
void snn_jump_sde_kernel(const float* __restrict__ w,
                         const float* __restrict__ ic,
                         const float* __restrict__ v0,
                         const float* __restrict__ i0,
                         const float* __restrict__ s0,
                         const float* __restrict__ mu,
                         const float* __restrict__ sigma,
                         const float* __restrict__ noise,      // [T,S,N,2]
                         const float* __restrict__ exp_draws,  // [T,S,N]
                         float* __restrict__ out)              // [S,N,T,3]
{
    __shared__ float lds_spk[2 * ROWS * LDS_STRIDE];

    const int tid   = threadIdx.x;
    const int wave  = tid >> 5;          // 0..7  -> 16-neuron column tile
    const int lane  = tid & 31;
    const int l16   = lane & 15;
    const int hi    = lane >> 4;         // 0 or 1 (half-wave)
    const int n     = wave * 16 + l16;   // this lane's neuron (B/C/D column)
    const int klo   = hi * 2;            // K offset inside A/B fragment
    const int rBase = hi * 8;            // sample-row base in C/D fragment
    const int sBase = blockIdx.x * ROWS; // this WG's first sample

    const float mu1 = mu[0], mu2 = mu[1];
    const float sqDT = sqrtf(DT_F);
    const float sg00 = sigma[0], sg01 = sigma[1];
    const float sg10 = sigma[2], sg11 = sigma[3];
    const float icn  = ic[n];

    // Preload time-invariant B fragments of w: bw[kb] holds rows kb*4+klo, +1.
    v2f bw[32];
#pragma unroll
    for (int kb = 0; kb < 32; ++kb) {
        const int row = kb * 4 + klo;
        bw[kb].x = w[row * NN + n];
        bw[kb].y = w[(row + 1) * NN + n];
    }

    // State in C/D fragment layout: component r <-> sample sBase+rBase+r, neuron n.
    v8f vS, iS, sS;
#pragma unroll
    for (int r = 0; r < 8; ++r) {
        const int sg = sBase + rBase + r;
        vS[r] = v0[sg * NN + n];
        iS[r] = i0[sg * NN + n];
        sS[r] = s0[sg * NN + n];
    }

    const float* Abase = &lds_spk[l16 * LDS_STRIDE + klo];

    for (int t = 0; t < T_STEPS; ++t) {
        const int bufOff = (t & 1) * (ROWS * LDS_STRIDE);
        float ed[8], spk[8];

#pragma unroll
        for (int r = 0; r < 8; ++r) {
            const int sg = sBase + rBase + r;
            const int base = (t * S_TOT + sg) * NN + n;
            const v2f dw = *(const v2f*)(noise + base * 2);
            ed[r] = exp_draws[base];

            float vv = vS[r], ii = iS[r], ss = sS[r];
            // drift
            const float dv = mu1 * (ii + icn - vv);
            const float di = -mu2 * ii;
            const float ds = fmaxf(vv, 0.f) + log1pf(expf(-fabsf(vv))); // softplus
            // diffusion: (sigma @ dW), dW = noise * sqrt(DT)
            const float dwa = dw.x * sqDT, dwb = dw.y * sqDT;
            const float nv0 = dwa * sg00 + dwb * sg01;
            const float nv1 = dwa * sg10 + dwb * sg11;
            vv = vv + DT_F * dv + nv0;
            ii = ii + DT_F * di + nv1;
            ss = ss + DT_F * ds;
            const float sp = (ss >= 0.f) ? 1.f : 0.f;
            vS[r] = vv; iS[r] = ii; sS[r] = ss; spk[r] = sp;
            lds_spk[bufOff + (rBase + r) * LDS_STRIDE + n] = sp;
        }

        if (t + 1 < T_STEPS) {  // uniform branch: EXEC stays all-ones
            __builtin_prefetch(noise + (((t + 1) * S_TOT + sBase + rBase) * NN + n) * 2, 0, 0);
        }
        __syncthreads();

        // i += spk @ w  via exact-fp32 WMMA; 4 independent accumulator chains.
        v8f acc0 = iS;
        v8f acc1 = {0.f,0.f,0.f,0.f,0.f,0.f,0.f,0.f};
        v8f acc2 = acc1, acc3 = acc1;
        const float* A = Abase + bufOff;
#pragma unroll
        for (int kb = 0; kb < 32; kb += 4) {
            const v2f a0 = *(const v2f*)(A + (kb + 0) * 4);
            const v2f a1 = *(const v2f*)(A + (kb + 1) * 4);
            const v2f a2 = *(const v2f*)(A + (kb + 2) * 4);
            const v2f a3 = *(const v2f*)(A + (kb + 3) * 4);
            acc0 = __builtin_amdgcn_wmma_f32_16x16x4_f32(false, a0, false, bw[kb + 0], (short)0, acc0, false, false);
            acc1 = __builtin_amdgcn_wmma_f32_16x16x4_f32(false, a1, false, bw[kb + 1], (short)0, acc1, false, false);
            acc2 = __builtin_amdgcn_wmma_f32_16x16x4_f32(false, a2, false, bw[kb + 2], (short)0, acc2, false, false);
            acc3 = __builtin_amdgcn_wmma_f32_16x16x4_f32(false, a3, false, bw[kb + 3], (short)0, acc3, false, false);
        }
        iS = (acc0 + acc1) + (acc2 + acc3);

        // Reset + record (post-coupling i, post-reset v, redrawn s), [S,N,T,3].
#pragma unroll
        for (int r = 0; r < 8; ++r) {
            const bool sp = spk[r] != 0.f;
            const float vv = sp ? 0.f : vS[r];
            const float ss = sp ? -ed[r] : sS[r];
            vS[r] = vv; sS[r] = ss;
            const int sg = sBase + rBase + r;
            const int ob = ((sg * NN + n) * T_STEPS + t) * 3;
            out[ob + 0] = vv;
            out[ob + 1] = iS[r];
            out[ob + 2] = ss;
        }
    }
}

extern "C" void kernel_launch(void* const* d_in, const int* in_sizes, int n_in,
                              void* d_out, int out_size, void* d_ws, size_t ws_size,
                              hipStream_t stream) {
    const float* w         = (const float*)d_in[0];
    const float* ic        = (const float*)d_in[1];
    const float* v0        = (const float*)d_in[2];
    const float* i0        = (const float*)d_in[3];
    const float* s0        = (const float*)d_in[4];
    const float* mu        = (const float*)d_in[5];
    const float* sigma     = (const float*)d_in[6];
    const float* noise     = (const float*)d_in[7];
    const float* exp_draws = (const float*)d_in[8];
    float* out = (float*)d_out;

    // 4 workgroups x 16 samples; 8 waves per WG (one 16-neuron tile each).
    snn_jump_sde_kernel<<<dim3(S_TOT / ROWS), dim3(256), 0, stream>>>(
        w, ic, v0, i0, s0, mu, sigma, noise, exp_draws, out);
}